// RankingLoss_403726926226
// MI455X (gfx1250) — compile-verified
//
#include <hip/hip_runtime.h>
#include <math.h>

// Problem constants (fixed by the reference).
#define B_ROWS 256
#define N_COLS 65536
#define TB     256                 // threads per block (8 wave32)
#define TILE_F 1024                // floats per tile per array (TB * 4)
#define NTILE  (N_COLS / TILE_F)   // 64 tiles per row
#define DEPTH  8                   // async prefetch depth (tiles in flight)

// ---- CDNA5 async global->LDS path (guarded; falls back to direct loads) ----
#if defined(__HIP_DEVICE_COMPILE__) && defined(__gfx1250__) && \
    __has_builtin(__builtin_amdgcn_global_load_async_to_lds_b128) && \
    __has_builtin(__builtin_amdgcn_s_wait_asynccnt)
#define HAVE_ASYNC 1
#else
#define HAVE_ASYNC 0
#endif

#if HAVE_ASYNC
// Builtin signature (from clang diagnostic): param is pointer to
// "__attribute__((__vector_size__(4*sizeof(int)))) int" — i.e. int4 —
// with AS1 (global) for the source and AS3 (LDS) for the destination.
typedef int v4i_t __attribute__((vector_size(16)));
typedef __attribute__((address_space(1))) v4i_t g_v4i;
typedef __attribute__((address_space(3))) v4i_t l_v4i;
#define ASYNC_CP16(gp, lp) \
  __builtin_amdgcn_global_load_async_to_lds_b128((g_v4i*)(gp), (l_v4i*)(lp), 0, 0)
#endif

// Online logsumexp accumulator update: state (m, s) means lse = m + log(s).
__device__ __forceinline__ void lse_update(float& m, float& s, float x) {
  if (x > m) {
    s = s * __expf(m - x) + 1.0f;  // m == -1e30 start: expf(-1e30 - x) -> 0
    m = x;
  } else {
    s += __expf(x - m);
  }
}

// Process one (sim, label) element.
__device__ __forceinline__ void proc1(float s, float y,
                                      float& mp, float& sp,
                                      float& mn, float& sn,
                                      int& nneg, float* t10) {
  if (y > 0.5f) {                                  // positive
    float a  = fmaxf(0.8f - s, 0.0f);
    float lp = -a * (s - 0.8f) * 64.0f;            // == 64*relu(0.8-s)^2
    lse_update(mp, sp, lp);
  }
  if (y < 0.25f) {                                 // negative
    ++nneg;
    float a  = fmaxf(s - 0.2f, 0.0f);
    float ln = a * (s - 0.2f) * 64.0f;             // == 64*relu(s-0.2)^2
    lse_update(mn, sn, ln);
    if (s > t10[9]) {                              // sorted-desc insert, regs only
      float v = s;
#pragma unroll
      for (int k = 0; k < 10; ++k) {
        float hi = fmaxf(t10[k], v);
        v        = fminf(t10[k], v);
        t10[k]   = hi;
      }
    }
  }
}

__global__ __launch_bounds__(TB) void ranking_row_kernel(
    const float* __restrict__ sim, const float* __restrict__ lab,
    float* __restrict__ rowloss) {
  const int row = blockIdx.x;
  const int tid = threadIdx.x;
  const float* __restrict__ srow = sim + (size_t)row * N_COLS;
  const float* __restrict__ lrow = lab + (size_t)row * N_COLS;
  const float4* __restrict__ s4 = (const float4*)srow;
  const float4* __restrict__ y4 = (const float4*)lrow;

  float mp = -1e30f, sp = 0.0f;     // positives lse state
  float mn = -1e30f, sn = 0.0f;     // all-negatives lse state
  int nneg = 0;
  float t10[10];
#pragma unroll
  for (int k = 0; k < 10; ++k) t10[k] = -1e30f;    // matches NEG_INF fill

#if HAVE_ASYNC
  // Depth-8 async global->LDS pipeline: 16 outstanding B128 async ops per
  // wave (8 KB/wave, ~16 MB across 2048 waves) to cover HBM BW*latency.
  __shared__ float4 stg_s[DEPTH][TB];
  __shared__ float4 stg_y[DEPTH][TB];
#pragma unroll
  for (int p = 0; p < DEPTH; ++p) {
    ASYNC_CP16(&s4[p * TB + tid], &stg_s[p][tid]);
    ASYNC_CP16(&y4[p * TB + tid], &stg_y[p][tid]);
  }
  for (int t = 0; t < NTILE; ++t) {
    if (t + DEPTH < NTILE) {
      __builtin_amdgcn_s_wait_asynccnt(2 * (DEPTH - 1));  // tile t's 2 ops done
    } else {
      __builtin_amdgcn_s_wait_asynccnt(0);                // drain at the tail
    }
    __syncthreads();                       // all waves' tile t visible in LDS
    float4 a = stg_s[t & (DEPTH - 1)][tid];
    float4 b = stg_y[t & (DEPTH - 1)][tid];
    proc1(a.x, b.x, mp, sp, mn, sn, nneg, t10);
    proc1(a.y, b.y, mp, sp, mn, sn, nneg, t10);
    proc1(a.z, b.z, mp, sp, mn, sn, nneg, t10);
    proc1(a.w, b.w, mp, sp, mn, sn, nneg, t10);
    __syncthreads();                       // everyone done reading this buffer
    int nt = t + DEPTH;
    if (nt < NTILE) {
      ASYNC_CP16(&s4[nt * TB + tid], &stg_s[t & (DEPTH - 1)][tid]);
      ASYNC_CP16(&y4[nt * TB + tid], &stg_y[t & (DEPTH - 1)][tid]);
    }
  }
#else
  // Fallback: direct coalesced global_load_b128 streaming.
#pragma unroll 4
  for (int t = 0; t < NTILE; ++t) {
    float4 a = s4[t * TB + tid];
    float4 b = y4[t * TB + tid];
    proc1(a.x, b.x, mp, sp, mn, sn, nneg, t10);
    proc1(a.y, b.y, mp, sp, mn, sn, nneg, t10);
    proc1(a.z, b.z, mp, sp, mn, sn, nneg, t10);
    proc1(a.w, b.w, mp, sp, mn, sn, nneg, t10);
  }
#endif

  // ---------------- block-level merges (all fixed-order => deterministic) ---
  __shared__ float rm[TB], rs[TB];
  __shared__ int   rix[TB];
  __shared__ float res2[2];
  __shared__ int   cnt_sh;
  __shared__ float topv[TB * 10];
  __shared__ float gtop[10];

  // 1) positives lse
  rm[tid] = mp; rs[tid] = sp;
  __syncthreads();
  for (int off = TB / 2; off > 0; off >>= 1) {
    if (tid < off) {
      float m1 = rm[tid], m2 = rm[tid + off];
      float M  = fmaxf(m1, m2);
      rs[tid]  = rs[tid] * __expf(m1 - M) + rs[tid + off] * __expf(m2 - M);
      rm[tid]  = M;
    }
    __syncthreads();
  }
  if (tid == 0) res2[0] = (rs[0] > 0.0f) ? (rm[0] + __logf(rs[0])) : 0.0f;
  __syncthreads();

  // 2) all-negatives lse + negative count (fused trees)
  rm[tid] = mn; rs[tid] = sn; rix[tid] = nneg;
  __syncthreads();
  for (int off = TB / 2; off > 0; off >>= 1) {
    if (tid < off) {
      float m1 = rm[tid], m2 = rm[tid + off];
      float M  = fmaxf(m1, m2);
      rs[tid]  = rs[tid] * __expf(m1 - M) + rs[tid + off] * __expf(m2 - M);
      rm[tid]  = M;
      rix[tid] += rix[tid + off];
    }
    __syncthreads();
  }
  if (tid == 0) {
    res2[1] = (rs[0] > 0.0f) ? (rm[0] + __logf(rs[0])) : -1e30f;
    cnt_sh  = rix[0];
  }

  // 3) global top-10 of negative sims: 10 rounds of argmax + invalidation
#pragma unroll
  for (int k = 0; k < 10; ++k) topv[tid * 10 + k] = t10[k];
  for (int q = 0; q < 10; ++q) {
    __syncthreads();
    float bv = -3.0e30f;
    int   bi = tid * 10;
#pragma unroll
    for (int k = 0; k < 10; ++k) {
      float v = topv[tid * 10 + k];
      if (v > bv) { bv = v; bi = tid * 10 + k; }
    }
    rm[tid] = bv; rix[tid] = bi;
    __syncthreads();
    for (int off = TB / 2; off > 0; off >>= 1) {
      if (tid < off && rm[tid + off] > rm[tid]) {
        rm[tid] = rm[tid + off]; rix[tid] = rix[tid + off];
      }
      __syncthreads();
    }
    if (tid == 0) {
      gtop[q] = rm[0];
      topv[rix[0]] = -3.0e30f;   // invalidate winner (below NEG_INF fill)
    }
  }
  __syncthreads();

  // 4) per-row loss (thread 0 only; all inputs were produced by thread 0)
  if (tid == 0) {
    float lse_p   = res2[0];
    float lse_all = res2[1];
    float L[10];
    float M = -3.0e30f;
#pragma unroll
    for (int q = 0; q < 10; ++q) {
      float x = gtop[q];
      float a = fmaxf(x - 0.2f, 0.0f);
      L[q]    = a * (x - 0.2f) * 64.0f;
      M       = fmaxf(M, L[q]);
    }
    float S = 0.0f;
#pragma unroll
    for (int q = 0; q < 10; ++q) S += __expf(L[q] - M);
    float lse_top = M + __logf(S);
    float lse_n   = (cnt_sh > 20) ? lse_top : lse_all;
    float z       = lse_n + lse_p;
    // stable softplus: max(z,0) + log1p(exp(-|z|))
    rowloss[row]  = fmaxf(z, 0.0f) + log1pf(__expf(-fabsf(z)));
  }
}

// Deterministic fixed-order mean over the 256 per-row losses.
__global__ __launch_bounds__(TB) void ranking_finalize_kernel(
    const float* __restrict__ rowloss, float* __restrict__ out) {
  __shared__ float sh[TB];
  const int tid = threadIdx.x;
  sh[tid] = rowloss[tid];
  __syncthreads();
  for (int off = TB / 2; off > 0; off >>= 1) {
    if (tid < off) sh[tid] += sh[tid + off];
    __syncthreads();
  }
  if (tid == 0) out[0] = sh[0] / (float)B_ROWS;
}

extern "C" void kernel_launch(void* const* d_in, const int* in_sizes, int n_in,
                              void* d_out, int out_size, void* d_ws, size_t ws_size,
                              hipStream_t stream) {
  (void)in_sizes; (void)n_in; (void)out_size; (void)ws_size;
  const float* sim = (const float*)d_in[0];
  const float* lab = (const float*)d_in[1];
  float* out     = (float*)d_out;
  float* rowloss = (float*)d_ws;   // B_ROWS floats of scratch

  ranking_row_kernel<<<B_ROWS, TB, 0, stream>>>(sim, lab, rowloss);
  ranking_finalize_kernel<<<1, TB, 0, stream>>>(rowloss, out);
}